// GlobalNGramEncoder_38208029065812
// MI455X (gfx1250) — compile-verified
//
#include <hip/hip_runtime.h>

#define VOCAB 33
#define V2 1089          // 33*33
#define BUCKETS 512
#define B_ROWS 4096
#define S_LEN 4096
#define KPAD_BI 1104     // 1089 padded to multiple of 16 (zero-filled)
#define HID 256
#define OUTC 64

typedef float v2f __attribute__((ext_vector_type(2)));
typedef float v8f __attribute__((ext_vector_type(8)));

// ---------------------------------------------------------------------------
// Kernel 1: per-row masked bigram/trigram histograms + normalization.
// One workgroup per row; LDS counters; LDS atomics; coalesced row reads.
// ---------------------------------------------------------------------------
__global__ __launch_bounds__(256) void hist_kernel(const int* __restrict__ src,
                                                   float* __restrict__ bi_freq,
                                                   float* __restrict__ tri_freq) {
    __shared__ unsigned bi_cnt[V2];
    __shared__ unsigned tri_cnt[BUCKETS];
    __shared__ unsigned totals[2];

    const int row = blockIdx.x;
    const int tid = threadIdx.x;
    const int* s = src + (size_t)row * S_LEN;

    for (int i = tid; i < V2; i += 256) bi_cnt[i] = 0u;
    for (int i = tid; i < BUCKETS; i += 256) tri_cnt[i] = 0u;
    if (tid < 2) totals[tid] = 0u;
    __syncthreads();

    unsigned myBi = 0, myTri = 0;
    for (int i = tid; i < S_LEN - 1; i += 256) {
        int c = s[i];
        int n = s[i + 1];
        bool bm = (c > 0) && (n > 0);
        if (bm) {
            atomicAdd(&bi_cnt[c * VOCAB + n], 1u);
            ++myBi;
        }
        if (i < S_LEN - 2) {
            int t = s[i + 2];
            if (bm && (t > 0)) {
                unsigned idx = ((unsigned)(c * (VOCAB * VOCAB) + n * VOCAB + t)) & (BUCKETS - 1);
                atomicAdd(&tri_cnt[idx], 1u);
                ++myTri;
            }
        }
    }
    if (myBi)  atomicAdd(&totals[0], myBi);
    if (myTri) atomicAdd(&totals[1], myTri);
    __syncthreads();

    const float biInv  = 1.0f / (float)max(totals[0], 1u);
    const float triInv = 1.0f / (float)max(totals[1], 1u);

    float* bf = bi_freq + (size_t)row * KPAD_BI;
    for (int i = tid; i < KPAD_BI; i += 256)
        bf[i] = (i < V2 ? (float)bi_cnt[i] : 0.0f) * biInv;

    float* tf = tri_freq + (size_t)row * BUCKETS;
    for (int i = tid; i < BUCKETS; i += 256)
        tf[i] = (float)tri_cnt[i] * triInv;
}

// ---------------------------------------------------------------------------
// Kernel 2: fused 2-layer MLP on a 16-row tile using V_WMMA_F32_16X16X4_F32.
//   hidden = relu(freq @ W1 + b1)   (K -> 256), 8 waves x 32 hidden cols
//   out    = hidden @ W2 + b2       (256 -> 32), 2 waves x 16 out cols
// A-fragment layout (f32 16x4): lanes 0-15 hold {K0,K1}, lanes 16-31 {K2,K3}.
// B mirrored (column n per lane, same K mapping). C/D: vgpr r, lane l ->
// (M = r + 8*(l>>4), N = l&15).
// ---------------------------------------------------------------------------
__global__ __launch_bounds__(256) void mlp_kernel(const float* __restrict__ freq,
                                                  int K, int Kpad,
                                                  const float* __restrict__ W1,
                                                  const float* __restrict__ b1,
                                                  const float* __restrict__ W2,
                                                  const float* __restrict__ b2,
                                                  float* __restrict__ out,
                                                  int outOff) {
    __shared__ float lds_h[16 * HID];   // 16 KB hidden tile

    const int rowBase = blockIdx.x * 16;
    const int lane = threadIdx.x & 31;
    const int wave = threadIdx.x >> 5;
    const int m  = lane & 15;           // row within tile (A) / col (B)
    const int kh = lane >> 4;           // K half-select: 0 -> K0/K1, 1 -> K2/K3
    const int n  = lane & 15;
    const int colBase = wave * 32;      // this wave's hidden-column base

    v8f c0 = {};
    v8f c1 = {};

    const float* arow = freq + (size_t)(rowBase + m) * Kpad;

    for (int k0 = 0; k0 < Kpad; k0 += 4) {
        const int ka = k0 + 2 * kh;
        v2f a;
        a.x = arow[ka];
        a.y = arow[ka + 1];
        // clamp B K-index: padded A rows are zero, so clamped B values are inert
        const int kb0 = min(ka,     K - 1);
        const int kb1 = min(ka + 1, K - 1);
        v2f bA, bB;
        bA.x = W1[(size_t)kb0 * HID + colBase + n];
        bA.y = W1[(size_t)kb1 * HID + colBase + n];
        bB.x = W1[(size_t)kb0 * HID + colBase + 16 + n];
        bB.y = W1[(size_t)kb1 * HID + colBase + 16 + n];
        c0 = __builtin_amdgcn_wmma_f32_16x16x4_f32(false, a, false, bA, (short)0, c0, false, false);
        c1 = __builtin_amdgcn_wmma_f32_16x16x4_f32(false, a, false, bB, (short)0, c1, false, false);
    }

    // bias + relu in C-layout, stage hidden tile to LDS
    #pragma unroll
    for (int r = 0; r < 8; ++r) {
        const int mm = r + 8 * kh;
        float h0 = c0[r] + b1[colBase + n];
        float h1 = c1[r] + b1[colBase + 16 + n];
        h0 = h0 > 0.0f ? h0 : 0.0f;
        h1 = h1 > 0.0f ? h1 : 0.0f;
        lds_h[mm * HID + colBase + n]      = h0;
        lds_h[mm * HID + colBase + 16 + n] = h1;
    }
    __syncthreads();

    // layer 2: 16x256 @ 256x32 on waves 0 and 1 (wave-uniform branch, EXEC all 1s)
    if (wave < 2) {
        const int cb = wave * 16;
        v8f d = {};
        for (int k0 = 0; k0 < HID; k0 += 4) {
            const int ka = k0 + 2 * kh;
            v2f a;
            a.x = lds_h[m * HID + ka];
            a.y = lds_h[m * HID + ka + 1];
            v2f b;
            b.x = W2[(size_t)ka * 32 + cb + n];
            b.y = W2[(size_t)(ka + 1) * 32 + cb + n];
            d = __builtin_amdgcn_wmma_f32_16x16x4_f32(false, a, false, b, (short)0, d, false, false);
        }
        #pragma unroll
        for (int r = 0; r < 8; ++r) {
            const int mm = r + 8 * kh;
            out[(size_t)(rowBase + mm) * OUTC + outOff + cb + n] = d[r] + b2[cb + n];
        }
    }
}

// ---------------------------------------------------------------------------
extern "C" void kernel_launch(void* const* d_in, const int* in_sizes, int n_in,
                              void* d_out, int out_size, void* d_ws, size_t ws_size,
                              hipStream_t stream) {
    const int*   src = (const int*)d_in[0];
    const float* Wb1 = (const float*)d_in[1];
    const float* bb1 = (const float*)d_in[2];
    const float* Wb2 = (const float*)d_in[3];
    const float* bb2 = (const float*)d_in[4];
    const float* Wt1 = (const float*)d_in[5];
    const float* bt1 = (const float*)d_in[6];
    const float* Wt2 = (const float*)d_in[7];
    const float* bt2 = (const float*)d_in[8];
    float* out = (float*)d_out;

    float* bi_freq  = (float*)d_ws;                               // 4096 x 1104
    float* tri_freq = bi_freq + (size_t)B_ROWS * KPAD_BI;         // 4096 x 512

    hist_kernel<<<B_ROWS, 256, 0, stream>>>(src, bi_freq, tri_freq);

    mlp_kernel<<<B_ROWS / 16, 256, 0, stream>>>(bi_freq, V2, KPAD_BI,
                                                Wb1, bb1, Wb2, bb2, out, 0);
    mlp_kernel<<<B_ROWS / 16, 256, 0, stream>>>(tri_freq, BUCKETS, BUCKETS,
                                                Wt1, bt1, Wt2, bt2, out, 32);
}